// RealSHT_39298950759022
// MI455X (gfx1250) — compile-verified
//
#include <hip/hip_runtime.h>
#include <hip/hip_bf16.h>
#include <math.h>
#include <stdint.h>

// ---------------------------------------------------------------------------
// RealSHT forward on MI455X (gfx1250):
//   Stage 0: build scaled DFT cos/sin matrices (tiny, per-launch, ws)
//   Stage 1: rfft-as-GEMM via V_WMMA_F32_16X16X4_F32, x-tile staged in LDS
//            via async global->LDS (ASYNCcnt), shared by 12 waves
//            (6 m-strips x {re,im}) -> x read once from HBM.
//   Stage 2: per-(m,comp) spectral GEMM over latitude, W-tile staged in LDS
//            via async global->LDS, shared by 8 waves (all 256 batch rows)
//            -> W read once from HBM.
// fp32 WMMA throughout to match the fp32 reference numerics.
// ---------------------------------------------------------------------------

typedef __attribute__((ext_vector_type(2))) float v2f;
typedef __attribute__((ext_vector_type(8))) float v8f;
// 8-byte vector loads from rows with odd float strides are only 4B-aligned:
typedef float v2fa __attribute__((ext_vector_type(2), aligned(4)));

constexpr int BATCH = 4, CH = 64, NLAT = 361, NLON = 720;
constexpr int MMAX = 361, LMAX = 360;
constexpr int BC    = BATCH * CH;          // 256
constexpr int NROWS = BC * NLAT;           // 92416 signal rows
constexpr int MPAD  = 384;                 // m padded to 6 strips of 64
constexpr int XPAD  = 724;                 // LDS row stride for x tile (bank-safe)
constexpr int KPAD  = 364;                 // LDS row stride for W tile (bank-safe)
constexpr long long FPLANE  = (long long)MPAD * NLON;   // floats per cos/sin plane
constexpr long long XFPLANE = (long long)MMAX * NROWS;  // floats per re/im plane

__device__ __forceinline__ v8f wmma4(v2f a, v2f b, v8f c) {
  // (neg_a, A, neg_b, B, c_mod, C, reuse_a, reuse_b)
  return __builtin_amdgcn_wmma_f32_16x16x4_f32(false, a, false, b, (short)0, c,
                                               false, false);
}

// Async global->LDS copy primitives (CDNA5 VGLOBAL async ops, ASYNCcnt).
// lds_off: wave-relative LDS byte offset (low 32 bits of a flat LDS pointer).
__device__ __forceinline__ void async_g2l_b32(unsigned lds_off, const float* g) {
  asm volatile("global_load_async_to_lds_b32 %0, %1, off"
               :: "v"(lds_off), "v"(g) : "memory");
}
__device__ __forceinline__ void async_g2l_b128(unsigned lds_off, const float* g) {
  asm volatile("global_load_async_to_lds_b128 %0, %1, off"
               :: "v"(lds_off), "v"(g) : "memory");
}
__device__ __forceinline__ void wait_async0() {
  asm volatile("s_wait_asynccnt 0x0" ::: "memory");
}
__device__ __forceinline__ unsigned lds_offset(const float* p) {
  return (unsigned)(uintptr_t)p;   // flat LDS pointer: low 32 bits = LDS offset
}

// --------------------------- Stage 0: DFT matrices -------------------------
// F layout: [cos | sin] planes, each [MPAD][NLON] (m-major, K contiguous).
// Scale 2*pi/720 baked in; angle reduced via (m*n mod 720) in double.
__global__ void init_dft(float* __restrict__ F) {
  int idx = blockIdx.x * blockDim.x + threadIdx.x;
  if (idx >= (int)FPLANE) return;
  int m = idx / NLON;
  int n = idx % NLON;
  float c = 0.f, s = 0.f;
  if (m < MMAX) {
    const double TWO_PI = 6.283185307179586476925286766559;
    double ang   = -TWO_PI * (double)((m * n) % NLON) / (double)NLON;
    double scale =  TWO_PI / (double)NLON;
    c = (float)(scale * cos(ang));
    s = (float)(scale * sin(ang));
  }
  F[idx]          = c;
  F[FPLANE + idx] = s;
}

// --------------------------- Stage 1: DFT GEMM -----------------------------
// One block = 12 waves = one 32-row x tile (in LDS) x all (6 m-strips, re/im).
// Per wave: 32(rows) x 64(m) tile, K=720.  A from LDS, B from F (L2-resident).
// Output layout xf[comp][m][bc*NLAT + lat] so stage-2 A tiles are K-contig.
__global__ void __launch_bounds__(384)
dft_wmma(const float* __restrict__ x, const float* __restrict__ F,
         float* __restrict__ xf) {
  __shared__ float xs[32 * XPAD];          // ~91 KB of the WGP's 320 KB LDS
  const int tid  = threadIdx.x;
  const int lane = tid & 31;
  const int wid  = tid >> 5;               // 0..11
  const int lr   = lane & 15;
  const int half = lane >> 4;              // K=0,1 vs K=2,3 sub-tile
  const int r0   = blockIdx.x * 32;
  const int comp   = wid / 6;              // 0 = cos(re), 1 = sin(im)
  const int mstrip = wid % 6;
  const int n0     = mstrip * 64;

  // Async, coalesced 32x720 x-tile stage-in (rows are 16B aligned -> b128).
  for (int r = wid; r < 32; r += 12) {
    const float* src = x + (size_t)(r0 + r) * NLON;
    for (int n4 = lane; n4 < NLON / 4; n4 += 32)
      async_g2l_b128(lds_offset(&xs[r * XPAD + n4 * 4]), src + n4 * 4);
  }
  wait_async0();
  __syncthreads();

  const float* Fp   = F  + (size_t)comp * FPLANE;
  float*       outp = xf + (size_t)comp * XFPLANE;

  const float* a0p = &xs[(size_t)lr * XPAD + half * 2];        // rows 0..15
  const float* a1p = a0p + 16 * XPAD;                          // rows 16..31
  const float* brow0 = Fp + (size_t)(n0 +  0 + lr) * NLON + half * 2;
  const float* brow1 = Fp + (size_t)(n0 + 16 + lr) * NLON + half * 2;
  const float* brow2 = Fp + (size_t)(n0 + 32 + lr) * NLON + half * 2;
  const float* brow3 = Fp + (size_t)(n0 + 48 + lr) * NLON + half * 2;

  v8f acc[2][4] = {};
#pragma unroll 4
  for (int k = 0; k < NLON; k += 4) {
    v2f a0 = *(const v2fa*)(a0p + k);      // ds_load, conflict-free stride
    v2f a1 = *(const v2fa*)(a1p + k);
    v2f b0 = *(const v2fa*)(brow0 + k);
    v2f b1 = *(const v2fa*)(brow1 + k);
    v2f b2 = *(const v2fa*)(brow2 + k);
    v2f b3 = *(const v2fa*)(brow3 + k);
    acc[0][0] = wmma4(a0, b0, acc[0][0]);
    acc[0][1] = wmma4(a0, b1, acc[0][1]);
    acc[0][2] = wmma4(a0, b2, acc[0][2]);
    acc[0][3] = wmma4(a0, b3, acc[0][3]);
    acc[1][0] = wmma4(a1, b0, acc[1][0]);
    acc[1][1] = wmma4(a1, b1, acc[1][1]);
    acc[1][2] = wmma4(a1, b2, acc[1][2]);
    acc[1][3] = wmma4(a1, b3, acc[1][3]);
  }

  // C layout: VGPR v holds (M = half*8 + v, N = lr); a lane's 8 values hit 8
  // consecutive rows of the [m][row] plane -> vector v8f store (2x b128).
#pragma unroll
  for (int j = 0; j < 4; ++j) {
    int m = n0 + j * 16 + lr;
    if (m < MMAX) {
      float* o = outp + (size_t)m * NROWS + r0 + half * 8;     // 32B aligned
      *(v8f*)(o)      = acc[0][j];
      *(v8f*)(o + 16) = acc[1][j];
    }
  }
}

// ----------------------- Stage 2: spectral contraction ---------------------
// One block = 8 waves = all 256 bc rows x one 64-wide l strip for one (m,comp).
// W tile (64 x KPAD, zero-padded in l and k) staged once in LDS via async
// loads, shared by all waves.  Grid ordered (l-strip fastest) so the 6 blocks
// per m reuse the 370 KB xf[m] plane in L2.
__global__ void __launch_bounds__(256)
spec_wmma(const float* __restrict__ xf, const float* __restrict__ w,
          float* __restrict__ out) {
  __shared__ float bs[64 * KPAD];          // ~93 KB
  const int tid  = threadIdx.x;
  const int lane = tid & 31;
  const int wid  = tid >> 5;               // 0..7 -> bc strip
  const int lr   = lane & 15;
  const int half = lane >> 4;
  const int l0   = blockIdx.x * 64;
  const int m    = blockIdx.y >> 1;
  const int comp = blockIdx.y & 1;
  const int bc0  = wid * 32;

  // Stage W tile: valid region async-copied (rows only 4B aligned -> b32);
  // pad regions zeroed with ordinary LDS stores (disjoint, ordered by the
  // async-wait + barrier below).
  const float* Wm = w + (size_t)m * LMAX * NLAT;
  for (int r = wid; r < 64; r += 8) {
    int l = l0 + r;
    if (l < LMAX) {
      if (lane < KPAD - NLAT) bs[r * KPAD + NLAT + lane] = 0.f;   // k pad
      const float* src = Wm + (size_t)l * NLAT;
      for (int n = lane; n < NLAT; n += 32)
        async_g2l_b32(lds_offset(&bs[r * KPAD + n]), src + n);
    } else {
      for (int n = lane; n < KPAD; n += 32) bs[r * KPAD + n] = 0.f; // l pad
    }
  }
  wait_async0();
  __syncthreads();

  const float* A     = xf + (size_t)comp * XFPLANE + (size_t)m * NROWS;
  const float* arow0 = A + (size_t)(bc0 + lr) * NLAT + half * 2;
  const float* arow1 = arow0 + (size_t)16 * NLAT;
  const float* b0p = &bs[(size_t)(lr)      * KPAD + half * 2];
  const float* b1p = b0p + 16 * KPAD;
  const float* b2p = b0p + 32 * KPAD;
  const float* b3p = b0p + 48 * KPAD;
  const v2f zero2 = {0.f, 0.f};

  v8f acc[2][4] = {};
  int k = 0;
#pragma unroll 4
  for (; k + 4 <= NLAT; k += 4) {          // 90 full K-steps
    v2f a0 = *(const v2fa*)(arow0 + k);
    v2f a1 = *(const v2fa*)(arow1 + k);
    v2f b0 = *(const v2fa*)(b0p + k);
    v2f b1 = *(const v2fa*)(b1p + k);
    v2f b2 = *(const v2fa*)(b2p + k);
    v2f b3 = *(const v2fa*)(b3p + k);
    acc[0][0] = wmma4(a0, b0, acc[0][0]);
    acc[0][1] = wmma4(a0, b1, acc[0][1]);
    acc[0][2] = wmma4(a0, b2, acc[0][2]);
    acc[0][3] = wmma4(a0, b3, acc[0][3]);
    acc[1][0] = wmma4(a1, b0, acc[1][0]);
    acc[1][1] = wmma4(a1, b1, acc[1][1]);
    acc[1][2] = wmma4(a1, b2, acc[1][2]);
    acc[1][3] = wmma4(a1, b3, acc[1][3]);
  }
  { // K tail (k = 360): LDS side is zero-padded; guard only the A side.
    v2f a0 = zero2, a1 = zero2;
    if (half == 0) {                       // element 360 valid, 361+ zero
      a0.x = arow0[k];
      a1.x = arow1[k];
    }
    v2f b0 = *(const v2fa*)(b0p + k);
    v2f b1 = *(const v2fa*)(b1p + k);
    v2f b2 = *(const v2fa*)(b2p + k);
    v2f b3 = *(const v2fa*)(b3p + k);
    acc[0][0] = wmma4(a0, b0, acc[0][0]);
    acc[0][1] = wmma4(a0, b1, acc[0][1]);
    acc[0][2] = wmma4(a0, b2, acc[0][2]);
    acc[0][3] = wmma4(a0, b3, acc[0][3]);
    acc[1][0] = wmma4(a1, b0, acc[1][0]);
    acc[1][1] = wmma4(a1, b1, acc[1][1]);
    acc[1][2] = wmma4(a1, b2, acc[1][2]);
    acc[1][3] = wmma4(a1, b3, acc[1][3]);
  }

  // out is complex64 interleaved: ((bc*LMAX + l)*MMAX + m)*2 + comp
#pragma unroll
  for (int j = 0; j < 4; ++j) {
    int l = l0 + j * 16 + lr;
    if (l < LMAX) {
#pragma unroll
      for (int i = 0; i < 2; ++i) {
#pragma unroll
        for (int v = 0; v < 8; ++v) {
          int bc = bc0 + i * 16 + half * 8 + v;
          out[(((size_t)bc * LMAX + l) * MMAX + m) * 2 + comp] = acc[i][j][v];
        }
      }
    }
  }
}

// ---------------------------------------------------------------------------
extern "C" void kernel_launch(void* const* d_in, const int* in_sizes, int n_in,
                              void* d_out, int out_size, void* d_ws, size_t ws_size,
                              hipStream_t stream) {
  const float* x = (const float*)d_in[0];   // (4,64,361,720) fp32
  const float* w = (const float*)d_in[1];   // (361,360,361) fp32
  float* out = (float*)d_out;               // (4,64,360,361) complex64 interleaved
  float* ws  = (float*)d_ws;

  // Workspace layout (floats): [Fc|Fs] then [XfRe|XfIm]  (~269 MB total)
  float* F  = ws;
  float* xf = ws + 2 * FPLANE;

  // Stage 0: DFT matrices
  init_dft<<<((int)FPLANE + 255) / 256, 256, 0, stream>>>(F);

  // Stage 1: one block per 32-row x tile; 12 waves cover (6 m-strips x re/im)
  dft_wmma<<<dim3(NROWS / 32), 384, 0, stream>>>(x, F, xf);

  // Stage 2: one block per (l-strip, m, comp); l-strip fastest for L2 reuse
  spec_wmma<<<dim3(MPAD / 64, MMAX * 2), 256, 0, stream>>>(xf, w, out);
}